// ADSR_38860864094844
// MI455X (gfx1250) — compile-verified
//
#include <hip/hip_runtime.h>

// ---------------------------------------------------------------------------
// ADSR envelope, exact per-run recurrence decomposed into 3 parallel kernels.
//   k_edges : per 2048-sample chunk, find gate edges. Gate tile is brought
//             into LDS via gfx1250 GLOBAL_LOAD_ASYNC_TO_LDS_B128 (ASYNCcnt).
//   k_chain : per row, walk runs sequentially, compute run start levels (tiny)
//   k_fill  : per sample, closed-form envelope from run-local info (parallel)
// ---------------------------------------------------------------------------

#define CHUNK      2048
#define MAXE       16
#define K_THREADS  256
#define SPT        (CHUNK / K_THREADS)   // 8 samples per thread

static_assert(CHUNK % K_THREADS == 0, "chunk must divide evenly");

#if defined(__gfx1250__) && __has_builtin(__builtin_amdgcn_global_load_async_to_lds_b128)
#define USE_ASYNC_LDS 1
typedef int v4i __attribute__((ext_vector_type(4)));
typedef __attribute__((address_space(1))) v4i* gbl_v4i_p;   // global (AS1) int4*
typedef __attribute__((address_space(3))) v4i* lds_v4i_p;   // LDS    (AS3) int4*
#else
#define USE_ASYNC_LDS 0
#endif

__global__ void __launch_bounds__(K_THREADS)
k_edges(const float* __restrict__ gate, int T, int nchpr,
        int* __restrict__ ecnt, int* __restrict__ epos)
{
    const int ch   = blockIdx.x;
    const int row  = ch / nchpr;
    const int cidx = ch % nchpr;
    const size_t base = (size_t)row * (size_t)T + (size_t)cidx * CHUNK;
    const float* g = gate + base;

    __shared__ float s_g[CHUNK];
    __shared__ int   cnt;
    if (threadIdx.x == 0) cnt = 0;

    const int tid = threadIdx.x;

    // warm L2 for the next streaming chunk (gfx1250 global_prefetch_b8)
    if (cidx + 1 < nchpr) __builtin_prefetch(g + tid * SPT + CHUNK, 0, 1);

#if USE_ASYNC_LDS
    // Async DMA-style copy of the 8KB gate tile into LDS (2 x b128 per lane),
    // tracked by ASYNCcnt -- no VGPR staging for the tile.
    __builtin_amdgcn_global_load_async_to_lds_b128(
        (gbl_v4i_p)(g + tid * 4),
        (lds_v4i_p)(s_g + tid * 4), 0, 0);
    __builtin_amdgcn_global_load_async_to_lds_b128(
        (gbl_v4i_p)(g + 1024 + tid * 4),
        (lds_v4i_p)(s_g + 1024 + tid * 4), 0, 0);
  #if __has_builtin(__builtin_amdgcn_s_wait_asynccnt)
    __builtin_amdgcn_s_wait_asynccnt(0);
  #else
    asm volatile("s_wait_asynccnt 0x0" ::: "memory");
  #endif
#else
    *(float4*)(s_g + tid * 4)        = *(const float4*)(g + tid * 4);
    *(float4*)(s_g + 1024 + tid * 4) = *(const float4*)(g + 1024 + tid * 4);
#endif
    __syncthreads();

    const int t0 = tid * SPT;
    const float4 v0 = *(const float4*)(s_g + t0);
    const float4 v1 = *(const float4*)(s_g + t0 + 4);
    float cur[SPT] = {v0.x, v0.y, v0.z, v0.w, v1.x, v1.y, v1.z, v1.w};

    float prev;
    if (t0 == 0) prev = (cidx == 0) ? v0.x   // row start: t=0 is never an "edge"
                                    : g[-1]; // last sample of previous chunk (same row)
    else         prev = s_g[t0 - 1];

    int* __restrict__ ep = epos + (size_t)ch * MAXE;
#pragma unroll
    for (int i = 0; i < SPT; ++i) {
        if (cur[i] != prev) {
            int k = atomicAdd(&cnt, 1);
            if (k < MAXE) ep[k] = t0 + i;       // chunk-relative position
        }
        prev = cur[i];
    }
    __syncthreads();
    if (threadIdx.x == 0) ecnt[ch] = (cnt < MAXE) ? cnt : MAXE;
}

// One thread per row: propagate (run_start, gate, start-level) across chunks,
// computing the start level of every new run, and per-chunk carry-in state.
__global__ void k_chain(const float* __restrict__ gate,
                        const float* __restrict__ att, const float* __restrict__ dcy,
                        const float* __restrict__ sus, const float* __restrict__ rel,
                        int T, int nchpr, int N,
                        const int* __restrict__ ecnt,
                        int* __restrict__ epos, float* __restrict__ elvl,
                        int* __restrict__ c_rs, int* __restrict__ c_g,
                        float* __restrict__ c_lvl)
{
    const int row = threadIdx.x;
    if (row >= N) return;

    const float a  = att[row];
    const float d  = dcy[row];
    const float su = sus[row];
    const float re = rel[row];

    int   g         = (gate[(size_t)row * (size_t)T] != 0.0f) ? 1 : 0;
    int   run_start = 0;       // within-row sample index where current run began
    float lvl       = 0.0f;    // on-run: a0 (attack start); off-run: r0 (release start)

    for (int c = 0; c < nchpr; ++c) {
        const int ch = row * nchpr + c;
        c_rs[ch]  = run_start;
        c_g[ch]   = g;
        c_lvl[ch] = lvl;

        const int cnt = ecnt[ch];
        int*   __restrict__ ep = epos + (size_t)ch * MAXE;
        float* __restrict__ el = elvl + (size_t)ch * MAXE;

        // atomics gave arbitrary order -> insertion sort (cnt <= 16, usually <= 1)
        for (int i = 1; i < cnt; ++i) {
            int v = ep[i], k = i;
            while (k > 0 && ep[k - 1] > v) { ep[k] = ep[k - 1]; --k; }
            ep[k] = v;
        }

        const int cbase = c * CHUNK;
        for (int i = 0; i < cnt; ++i) {
            const int   p = cbase + ep[i];            // new run's first sample
            const float L = (float)(p - run_start);   // closed run length; last axis == L
            float nl;
            if (g) {
                // closing an on-run: envelope at its last sample
                float e = (L <= a) ? (1.0f - lvl) * (L / a) + lvl
                                   : su + (1.0f - su) * __expf(-(L - a) / d);
                nl = (e == 0.0f) ? 1.0f : e;          // ri==0 -> 1 rule (release start)
            } else {
                // closing an off-run: release level at its last sample (a0 of next attack)
                nl = lvl * __expf(-L / re);           // lvl==0 before first note -> stays 0
            }
            el[i]     = nl;
            lvl       = nl;
            run_start = p;
            g        ^= 1;
        }
    }
}

__global__ void __launch_bounds__(K_THREADS)
k_fill(const float* __restrict__ att, const float* __restrict__ dcy,
       const float* __restrict__ sus, const float* __restrict__ rel,
       int T, int nchpr,
       const int* __restrict__ ecnt, const int* __restrict__ epos,
       const float* __restrict__ elvl,
       const int* __restrict__ c_rs, const int* __restrict__ c_g,
       const float* __restrict__ c_lvl,
       float* __restrict__ out)
{
    const int ch   = blockIdx.x;
    const int row  = ch / nchpr;
    const int cidx = ch % nchpr;

    __shared__ int   s_pos[MAXE];
    __shared__ float s_lvl[MAXE];
    __shared__ int   s_cnt, s_rs, s_g;
    __shared__ float s_l0, s_a, s_d, s_su, s_re;

    if (threadIdx.x == 0) {
        s_cnt = ecnt[ch];
        s_rs  = c_rs[ch];
        s_g   = c_g[ch];
        s_l0  = c_lvl[ch];
        s_a   = att[row]; s_d = dcy[row]; s_su = sus[row]; s_re = rel[row];
    }
    if (threadIdx.x < MAXE) {
        s_pos[threadIdx.x] = epos[(size_t)ch * MAXE + threadIdx.x];
        s_lvl[threadIdx.x] = elvl[(size_t)ch * MAXE + threadIdx.x];
    }
    __syncthreads();

    const int   cnt = s_cnt;
    const float a  = s_a, su = s_su;
    // per-row constants: 3 real divisions per thread instead of up to 3/sample
    const float inv_a  = 1.0f / a;
    const float inv_d  = 1.0f / s_d;
    const float inv_re = 1.0f / s_re;
    const float c_dec  = 1.0f - su;
    const int   trel0  = threadIdx.x * SPT;

    int   j = 0;                 // #edges at positions <= current sample
    float o[SPT];
#pragma unroll
    for (int i = 0; i < SPT; ++i) {
        const int tr = trel0 + i;
        while (j < cnt && s_pos[j] <= tr) ++j;

        int   g;
        float lvl, k;
        if (j == 0) {            // still inside the run carried in from previous chunks
            g   = s_g;
            lvl = s_l0;
            k   = (float)(cidx * CHUNK + tr - s_rs + 1);
        } else {                 // run started at edge j-1 inside this chunk
            g   = s_g ^ (j & 1);
            lvl = s_lvl[j - 1];
            k   = (float)(tr - s_pos[j - 1] + 1);
        }

        float env;
        if (g) {
            env = (k <= a) ? (1.0f - lvl) * (k * inv_a) + lvl             // attack ramp from a0
                           : su + c_dec * __expf((a - k) * inv_d);        // analytic decay
        } else {
            env = lvl * __expf(-k * inv_re);                              // release (0 pre-note)
        }
        o[i] = env;
    }

    float4* o4 = (float4*)(out + (size_t)row * (size_t)T + (size_t)cidx * CHUNK + trel0);
    o4[0] = make_float4(o[0], o[1], o[2], o[3]);
    o4[1] = make_float4(o[4], o[5], o[6], o[7]);
}

// ---------------------------------------------------------------------------

static inline size_t align16(size_t x) { return (x + 15) & ~(size_t)15; }

extern "C" void kernel_launch(void* const* d_in, const int* in_sizes, int n_in,
                              void* d_out, int out_size, void* d_ws, size_t ws_size,
                              hipStream_t stream)
{
    const float* gate = (const float*)d_in[0];
    const float* att  = (const float*)d_in[1];
    const float* dcy  = (const float*)d_in[2];
    const float* sus  = (const float*)d_in[3];
    const float* rel  = (const float*)d_in[4];

    const int N     = in_sizes[1];            // attack is [N,1]
    const int T     = in_sizes[0] / N;
    const int nchpr = T / CHUNK;
    const int nch   = N * nchpr;

    char* w = (char*)d_ws;
    int*   ecnt = (int*)w;   w += align16((size_t)nch * sizeof(int));
    int*   epos = (int*)w;   w += align16((size_t)nch * MAXE * sizeof(int));
    float* elvl = (float*)w; w += align16((size_t)nch * MAXE * sizeof(float));
    int*   crs  = (int*)w;   w += align16((size_t)nch * sizeof(int));
    int*   cg   = (int*)w;   w += align16((size_t)nch * sizeof(int));
    float* clv  = (float*)w; w += align16((size_t)nch * sizeof(float));

    k_edges<<<dim3(nch), dim3(K_THREADS), 0, stream>>>(gate, T, nchpr, ecnt, epos);

    k_chain<<<dim3(1), dim3(((N + 31) / 32) * 32), 0, stream>>>(
        gate, att, dcy, sus, rel, T, nchpr, N, ecnt, epos, elvl, crs, cg, clv);

    k_fill<<<dim3(nch), dim3(K_THREADS), 0, stream>>>(
        att, dcy, sus, rel, T, nchpr, ecnt, epos, elvl, crs, cg, clv, (float*)d_out);
}